// SelectiveSSM_33243046871751
// MI455X (gfx1250) — compile-verified
//
#include <hip/hip_runtime.h>
#include <math.h>

// Problem constants (from reference)
#define BB 2
#define TT 2048
#define DM 768
#define DI 1536
#define DS 16
#define DC 4
#define MBT (BB * TT)   // 4096 rows

typedef float v2f __attribute__((ext_vector_type(2)));
typedef float v8f __attribute__((ext_vector_type(8)));

__device__ __forceinline__ float silu_(float v) { return v / (1.f + __expf(-v)); }
__device__ __forceinline__ v2f ld2(const float* __restrict__ p) { v2f r; r.x = p[0]; r.y = p[1]; return r; }

#define WMMA_F32(a, b, c) \
    __builtin_amdgcn_wmma_f32_16x16x4_f32(false, (a), false, (b), (short)0, (c), false, false)

// ---------------------------------------------------------------------------
// WMMA fp32 GEMM:  C[M,N] = A[M,K] @ W[N,K]^T  (+ optional bias, softplus)
// One wave computes a 16x64 C strip (4 accumulators, A reuse x4); 4 waves per
// block -> 16x256 block tile.  Software-pipelined: the next 8-deep K chunk's
// fragments are loaded into a shadow register set while the current chunk's
// 8 WMMAs issue, so waits only cover the previous iteration's loads.
// Fragment layouts per ISA 7.12.2:
//   A 16x4 f32: lanes 0-15 -> M=lane, K={0,1}; lanes 16-31 -> K={2,3}
//   B mirrors A with M<->N (sourced from W[n,k] row-major, stride K)
//   C/D: VGPR i -> row m0+i (lanes 0-15) / m0+8+i (lanes 16-31), col=lane&15
// ---------------------------------------------------------------------------
template <int ACT>   // 0 = none, 1 = softplus(bias + v)
__global__ void gemm_wmma_f32(const float* __restrict__ A,
                              const float* __restrict__ W,
                              const float* __restrict__ bias,
                              float* __restrict__ C,
                              int M, int N, int K)
{
    const int lane = threadIdx.x & 31;
    const int wave = threadIdx.x >> 5;
    const int m0   = blockIdx.y * 16;
    const int n0   = blockIdx.x * 256 + wave * 64;
    const int row  = lane & 15;
    const int kk   = (lane >> 4) << 1;   // 0 or 2

    const float* ap = A + (size_t)(m0 + row) * K + kk;
    const float* wp[4];
    wp[0] = W + (size_t)(n0 + row) * K + kk;
    wp[1] = wp[0] + (size_t)16 * K;
    wp[2] = wp[0] + (size_t)32 * K;
    wp[3] = wp[0] + (size_t)48 * K;

    v8f acc[4] = { {}, {}, {}, {} };

    // prologue: load chunk 0 (k-steps 0 and 4)
    v2f a0 = ld2(ap), a1 = ld2(ap + 4);
    v2f b0[4], b1[4];
    #pragma unroll
    for (int j = 0; j < 4; ++j) { b0[j] = ld2(wp[j]); b1[j] = ld2(wp[j] + 4); }

    // pipelined main loop: prefetch chunk k0+8 while computing chunk k0
    int k0 = 0;
    for (; k0 + 8 < K; k0 += 8) {
        const int kn = k0 + 8;
        v2f na0 = ld2(ap + kn), na1 = ld2(ap + kn + 4);
        v2f nb0[4], nb1[4];
        #pragma unroll
        for (int j = 0; j < 4; ++j) { nb0[j] = ld2(wp[j] + kn); nb1[j] = ld2(wp[j] + kn + 4); }

        #pragma unroll
        for (int j = 0; j < 4; ++j) acc[j] = WMMA_F32(a0, b0[j], acc[j]);
        #pragma unroll
        for (int j = 0; j < 4; ++j) acc[j] = WMMA_F32(a1, b1[j], acc[j]);

        a0 = na0; a1 = na1;
        #pragma unroll
        for (int j = 0; j < 4; ++j) { b0[j] = nb0[j]; b1[j] = nb1[j]; }
    }
    // tail chunk (fragments already in registers)
    #pragma unroll
    for (int j = 0; j < 4; ++j) acc[j] = WMMA_F32(a0, b0[j], acc[j]);
    #pragma unroll
    for (int j = 0; j < 4; ++j) acc[j] = WMMA_F32(a1, b1[j], acc[j]);

    // epilogue
    const int col   = lane & 15;
    const int rbase = m0 + ((lane >> 4) << 3);
    #pragma unroll
    for (int i = 0; i < 8; ++i) {
        const int r = rbase + i;
        #pragma unroll
        for (int j = 0; j < 4; ++j) {
            const int c = n0 + j * 16 + col;
            float o = acc[j][i];
            if (bias) o += bias[c];
            if (ACT == 1) o = (o > 20.f) ? o : log1pf(__expf(o));
            C[(size_t)r * N + c] = o;
        }
    }
}

// ---------------------------------------------------------------------------
// Depthwise causal conv (width 4) + bias + SiLU.  Reads x_proj half of xz.
// ---------------------------------------------------------------------------
__global__ void conv_silu_kernel(const float* __restrict__ xz,
                                 const float* __restrict__ cw,
                                 const float* __restrict__ cb,
                                 float* __restrict__ xc)
{
    const int idx = blockIdx.x * blockDim.x + threadIdx.x;
    if (idx >= MBT * DI) return;
    const int d  = idx % DI;
    const int bt = idx / DI;
    const int t  = bt % TT;
    const int b  = bt / TT;
    float s = cb[d];
    #pragma unroll
    for (int j = 0; j < DC; ++j) {
        const int tt = t - (DC - 1) + j;
        if (tt >= 0)
            s += xz[((size_t)(b * TT + tt)) * (2 * DI) + d] * cw[d * DC + j];
    }
    xc[idx] = silu_(s);
}

// ---------------------------------------------------------------------------
// B/C projections (N = 16 each).  One block per (b,t) row; xc row staged in
// LDS, threads 0..15 -> Bv[n], threads 16..31 -> Cv[n].
// ---------------------------------------------------------------------------
__global__ void bc_proj_kernel(const float* __restrict__ xc,
                               const float* __restrict__ Bw,
                               const float* __restrict__ Cw,
                               float* __restrict__ Bv,
                               float* __restrict__ Cv)
{
    __shared__ float rowbuf[DI];
    const int bt = blockIdx.x;
    for (int d = threadIdx.x; d < DI; d += blockDim.x)
        rowbuf[d] = xc[(size_t)bt * DI + d];
    __syncthreads();
    if (threadIdx.x < 32) {
        const int n = threadIdx.x & 15;
        const float* Wn = (threadIdx.x < 16 ? Bw : Cw) + (size_t)n * DI;
        float s = 0.f;
        for (int d = 0; d < DI; ++d) s += rowbuf[d] * Wn[d];
        (threadIdx.x < 16 ? Bv : Cv)[bt * DS + n] = s;
    }
}

// ---------------------------------------------------------------------------
// Selective scan + gate.  One 16-lane group per (b,d) channel, lane = state n.
// h_n <- exp(dt*A_n)*h_n + dt*B_n*x ;  y_t = sum_n h_n*C_n (shfl_xor reduce).
// Fused epilogue: y = y*silu(z) + xc*D  (z taken from second half of xz).
// Latency-bound serial loop -> prefetch t+8 operand rows (global_prefetch_b8).
// ---------------------------------------------------------------------------
__global__ void scan_gate_kernel(const float* __restrict__ dt,
                                 const float* __restrict__ xc,
                                 const float* __restrict__ xz,
                                 const float* __restrict__ Bv,
                                 const float* __restrict__ Cv,
                                 const float* __restrict__ A_log,
                                 const float* __restrict__ Dp,
                                 float* __restrict__ y)
{
    const int gid = blockIdx.x * (blockDim.x >> 4) + (threadIdx.x >> 4);
    const int n   = threadIdx.x & 15;
    if (gid >= BB * DI) return;
    const int d = gid % DI;
    const int b = gid / DI;

    const float a_n = -__expf(A_log[d * DS + n]);
    const float Dd  = Dp[d];
    float h = 0.f;
    const size_t base = (size_t)b * TT;
    for (int t = 0; t < TT; ++t) {
        const size_t bt  = base + t;
        if (t + 8 < TT) {
            const size_t btp = bt + 8;
            __builtin_prefetch(dt + btp * DI + d, 0, 3);
            __builtin_prefetch(xc + btp * DI + d, 0, 3);
            __builtin_prefetch(Bv + btp * DS + n, 0, 3);
        }
        const float dtv  = dt[bt * DI + d];
        const float xv   = xc[bt * DI + d];
        const float bn   = Bv[bt * DS + n];
        const float cn   = Cv[bt * DS + n];
        h = __expf(dtv * a_n) * h + dtv * bn * xv;
        float p = h * cn;
        p += __shfl_xor(p, 8, 16);
        p += __shfl_xor(p, 4, 16);
        p += __shfl_xor(p, 2, 16);
        p += __shfl_xor(p, 1, 16);
        if (n == 0) {
            const float zv = xz[bt * (2 * DI) + DI + d];
            y[bt * DI + d] = p * silu_(zv) + xv * Dd;
        }
    }
}

// ---------------------------------------------------------------------------
// RMSNorm(out + x) * norm_w.  One block per (b,t) row of 768.
// ---------------------------------------------------------------------------
__global__ void rmsnorm_kernel(const float* __restrict__ og,
                               const float* __restrict__ x,
                               const float* __restrict__ nw,
                               float* __restrict__ out)
{
    __shared__ float rowbuf[DM];
    __shared__ float red[256];
    const int bt = blockIdx.x;
    float s = 0.f;
    for (int d = threadIdx.x; d < DM; d += blockDim.x) {
        const float v = og[(size_t)bt * DM + d] + x[(size_t)bt * DM + d];
        rowbuf[d] = v;
        s += v * v;
    }
    red[threadIdx.x] = s;
    __syncthreads();
    for (int off = 128; off > 0; off >>= 1) {
        if ((int)threadIdx.x < off) red[threadIdx.x] += red[threadIdx.x + off];
        __syncthreads();
    }
    const float scale = rsqrtf(red[0] / (float)DM + 1e-6f);
    for (int d = threadIdx.x; d < DM; d += blockDim.x)
        out[(size_t)bt * DM + d] = rowbuf[d] * scale * nw[d];
}

// ---------------------------------------------------------------------------
extern "C" void kernel_launch(void* const* d_in, const int* in_sizes, int n_in,
                              void* d_out, int out_size, void* d_ws, size_t ws_size,
                              hipStream_t stream)
{
    const float* x         = (const float*)d_in[0];
    const float* in_proj_w = (const float*)d_in[1];
    const float* conv_w    = (const float*)d_in[2];
    const float* conv_b    = (const float*)d_in[3];
    const float* dt_w      = (const float*)d_in[4];
    const float* dt_b      = (const float*)d_in[5];
    const float* A_log     = (const float*)d_in[6];
    const float* B_w       = (const float*)d_in[7];
    const float* C_w       = (const float*)d_in[8];
    const float* D_param   = (const float*)d_in[9];
    const float* out_w     = (const float*)d_in[10];
    const float* norm_w    = (const float*)d_in[11];

    float* ws = (float*)d_ws;
    float* xz = ws;                                   // 4096*3072
    float* xc = xz + (size_t)MBT * 2 * DI;            // 4096*1536
    float* dt = xc + (size_t)MBT * DI;                // 4096*1536
    float* Bv = dt + (size_t)MBT * DI;                // 4096*16
    float* Cv = Bv + (size_t)MBT * DS;                // 4096*16
    float* yb = Cv + (size_t)MBT * DS;                // 4096*1536
    float* og = yb + (size_t)MBT * DI;                // 4096*768

    // 1) xz = x @ in_proj_w^T  (M=4096, N=3072, K=768)
    {
        dim3 g(2 * DI / 256, MBT / 16);
        gemm_wmma_f32<0><<<g, 128, 0, stream>>>(x, in_proj_w, nullptr, xz, MBT, 2 * DI, DM);
    }
    // 2) depthwise conv + SiLU -> xc
    conv_silu_kernel<<<(MBT * DI + 255) / 256, 256, 0, stream>>>(xz, conv_w, conv_b, xc);
    // 3) dt = softplus(xc @ dt_w^T + dt_b)  (M=4096, N=1536, K=1536)
    {
        dim3 g(DI / 256, MBT / 16);
        gemm_wmma_f32<1><<<g, 128, 0, stream>>>(xc, dt_w, dt_b, dt, MBT, DI, DI);
    }
    // 4) Bv / Cv projections (N=16)
    bc_proj_kernel<<<MBT, 256, 0, stream>>>(xc, B_w, C_w, Bv, Cv);
    // 5) selective scan + gate -> yb
    scan_gate_kernel<<<(BB * DI) / 16, 256, 0, stream>>>(dt, xc, xz, Bv, Cv, A_log, D_param, yb);
    // 6) og = yb @ out_w^T  (M=4096, N=768, K=1536)
    {
        dim3 g(DM / 256, MBT / 16);
        gemm_wmma_f32<0><<<g, 128, 0, stream>>>(yb, out_w, nullptr, og, MBT, DM, DI);
    }
    // 7) rmsnorm(og + x)
    rmsnorm_kernel<<<MBT, 256, 0, stream>>>(og, x, norm_w, (float*)d_out);
}